// GatedConv2d_27668179321454
// MI455X (gfx1250) — compile-verified
//
#include <hip/hip_runtime.h>
#include <hip/hip_bf16.h>

// Gated mixture-of-experts conv2d (E=4, B=32, CIN=COUT=64, H=W=112, K=3, pad 1).
// Identity: out[b] = conv(x[b], sum_e gate[b,e]*W[e]) + sum_e gate[b,e]*bias[e].
// One implicit GEMM per batch: M=COUT=64, N=H*W (tiled 16), K=CIN*9=576,
// via V_WMMA_F32_16X16X4_F32 (fp32 in/acc == reference precision).
//
// Register tiling: each wave owns 2 output rows x 4 M-tiles (8 v8f accumulators).
// Per K-step: 4 A ds_load_b64 + 2 B ds_load_b64 feed 8 WMMAs (0.75 DS/WMMA).
// WMMA 16x16x4 f32 lane layout (ISA 7.12.2): lane supplies K = {2*half, 2*half+1}
// in VGPR0/1, so LDS is k-pair interleaved to make each fragment one b64 load.

#define E_EXP 4
#define BATCH 32
#define CIN   64
#define COUT  64
#define HH    112
#define WW    112

typedef __attribute__((ext_vector_type(2))) float v2f;
typedef __attribute__((ext_vector_type(8))) float v8f;

// ---- LDS partition (floats) ----
// wlds[r=9][p=ci/2 (32)][co (64)][2]     gated weights, k-pair interleaved  36864
// xlds[p=ci/2 (32)][row (6)][c (114)][2] padded input rows h0-1..h0+4       43776
// blds[64]                               gated bias                            64
#define W_LDS_FLOATS (9 * 32 * 64 * 2)
#define X_LDS_FLOATS (32 * 6 * 114 * 2)
#define B_LDS_FLOATS (64)
#define SMEM_FLOATS  (W_LDS_FLOATS + X_LDS_FLOATS + B_LDS_FLOATS) // 80704 -> 315.3 KB

#define ROWS_PER_WG 4                      // output rows per workgroup
#define XROWS       (ROWS_PER_WG + 2)      // staged input rows (halo)
#define NTILES      7                      // 7 * 16 = 112 columns
#define NWAVES      14                     // (row-pair) x (ntile)
#define BLOCK       (NWAVES * 32)          // 448 threads

__global__ void __launch_bounds__(BLOCK)
gated_conv_wmma_kernel(const float* __restrict__ x,
                       const float* __restrict__ layer_mask,
                       const float* __restrict__ W,
                       const float* __restrict__ bias,
                       float* __restrict__ out) {
    extern __shared__ float smem[];
    float* wlds = smem;                              // ((r*32+p)*64+co)*2 + j
    float* xlds = smem + W_LDS_FLOATS;               // ((p*6+row)*114+c)*2 + j
    float* blds = smem + W_LDS_FLOATS + X_LDS_FLOATS;

    const int h0  = blockIdx.x * ROWS_PER_WG;  // first output row of this WG
    const int b   = blockIdx.y;                // batch image
    const int tid = threadIdx.x;

    // ---- per-image gates ----
    const float g0 = layer_mask[b * E_EXP + 0];
    const float g1 = layer_mask[b * E_EXP + 1];
    const float g2 = layer_mask[b * E_EXP + 2];
    const float g3 = layer_mask[b * E_EXP + 3];

    // ---- stage padded input rows h0-1 .. h0+4, k-pair interleaved ----
    for (int idx = tid; idx < 32 * XROWS * 114; idx += BLOCK) {
        int p   = idx / (XROWS * 114);
        int rem = idx - p * (XROWS * 114);
        int row = rem / 114;
        int c   = rem - row * 114;
        int hh  = h0 + row - 1;
        int wwc = c - 1;
        bool inb = (hh >= 0) & (hh < HH) & (wwc >= 0) & (wwc < WW);
#pragma unroll
        for (int j = 0; j < 2; ++j) {
            int ci = p * 2 + j;
            float v = inb ? x[((b * CIN + ci) * HH + hh) * WW + wwc] : 0.0f;
            xlds[idx * 2 + j] = v;
        }
    }

    // ---- gated weight combine straight into LDS (coalesced W reads) ----
    for (int widx = tid; widx < 36864; widx += BLOCK) {
        int co  = widx / 576;
        int rem = widx - co * 576;
        int ci  = rem / 9;
        int r   = rem - ci * 9;          // kh*3+kw
        float acc = g0 * W[0 * 36864 + widx]
                  + g1 * W[1 * 36864 + widx]
                  + g2 * W[2 * 36864 + widx]
                  + g3 * W[3 * 36864 + widx];
        wlds[((r * 32 + (ci >> 1)) * 64 + co) * 2 + (ci & 1)] = acc;
    }

    if (tid < COUT) {
        blds[tid] = g0 * bias[0 * COUT + tid] + g1 * bias[1 * COUT + tid]
                  + g2 * bias[2 * COUT + tid] + g3 * bias[3 * COUT + tid];
    }
    __syncthreads();

    // ---- implicit GEMM: wave = (row pair, spatial tile); 2x4 block tile ----
    const int wave = tid >> 5;           // 0..13
    const int rp   = wave / NTILES;      // 0..1 : which pair of output rows
    const int nt   = wave - rp * NTILES; // 0..6 : 16-wide spatial tile
    const int lane = tid & 31;
    const int col  = lane & 15;
    const int half = lane >> 4;          // selects K pair {2h, 2h+1}
    const int w0   = nt * 16;
    const int r0   = rp * 2;             // first output row (block-local) of wave

    v8f acc[2][4] = {};                  // [row-in-pair][m-tile]

    for (int r = 0; r < 9; ++r) {
        const int kh = r / 3;
        const int kw = r - kh * 3;
        // base pointers (pair term cig*2 added in-loop; affine addressing)
        const float* xb = &xlds[((half * XROWS + (r0 + kh)) * 114 + (w0 + col + kw)) * 2];
        const float* wb = &wlds[((r * 32 + half) * 64 + col) * 2];
        for (int cig = 0; cig < 16; ++cig) {
            // 2 B fragments (two output rows), 1 ds_load_b64 each
            v2f bf0 = *(const v2f*)(xb + (cig * 2) * (XROWS * 114) * 2);
            v2f bf1 = *(const v2f*)(xb + (cig * 2) * (XROWS * 114) * 2 + 114 * 2);
#pragma unroll
            for (int m = 0; m < 4; ++m) {
                // 1 A fragment shared by both rows, 1 ds_load_b64
                v2f af = *(const v2f*)(wb + ((cig * 2) * 64 + m * 16) * 2);
                acc[0][m] = __builtin_amdgcn_wmma_f32_16x16x4_f32(
                                false, af, false, bf0, (short)0, acc[0][m], false, false);
                acc[1][m] = __builtin_amdgcn_wmma_f32_16x16x4_f32(
                                false, af, false, bf1, (short)0, acc[1][m], false, false);
            }
        }
    }

    // ---- epilogue: C/D layout (VGPR rr: M = rr + half*8, N = col) ----
#pragma unroll
    for (int roi = 0; roi < 2; ++roi) {
        const int h = h0 + r0 + roi;
#pragma unroll
        for (int m = 0; m < 4; ++m) {
#pragma unroll
            for (int rr = 0; rr < 8; ++rr) {
                int co = m * 16 + rr + half * 8;
                out[((b * COUT + co) * HH + h) * WW + w0 + col] = acc[roi][m][rr] + blds[co];
            }
        }
    }
}

extern "C" void kernel_launch(void* const* d_in, const int* in_sizes, int n_in,
                              void* d_out, int out_size, void* d_ws, size_t ws_size,
                              hipStream_t stream) {
    (void)in_sizes; (void)n_in; (void)d_ws; (void)ws_size; (void)out_size;
    const float* x    = (const float*)d_in[0];   // [B,CIN,H,W]
    const float* mask = (const float*)d_in[1];   // [B,E]
    const float* W    = (const float*)d_in[2];   // [E,COUT,CIN,3,3]
    const float* bias = (const float*)d_in[3];   // [E,COUT]
    float* out        = (float*)d_out;           // [B,COUT,H,W]

    dim3 grid(HH / ROWS_PER_WG, BATCH);          // (28, 32)
    dim3 block(BLOCK);                           // 448 = 14 wave32
    size_t smem_bytes = SMEM_FLOATS * sizeof(float); // 322,816 B <= 320 KB/WG limit
    gated_conv_wmma_kernel<<<grid, block, smem_bytes, stream>>>(x, mask, W, bias, out);
}